// GroupedQueryAttention_34849364640001
// MI455X (gfx1250) — compile-verified
//
#include <hip/hip_runtime.h>
#include <math.h>

// GroupedQueryAttention for gfx1250 (MI455X), fp32 end-to-end using
// V_WMMA_F32_16X16X4_F32 (wave32 WMMA), register-blocked for A/B-fragment
// reuse. B=1, S=2048, D=128, 32 Q heads, 8 KV heads (q head h -> kv head h%8
// per jnp.tile), RoPE on K and V, causal mask, softmax, output projection.

typedef __attribute__((ext_vector_type(2))) float v2f;
typedef __attribute__((ext_vector_type(8))) float v8f;

#define S_LEN 2048
#define DHEAD 128
#define NQH   32
#define NKV   8

__device__ __forceinline__ v8f wmma_f32(v2f a, v2f b, v8f c) {
    // D = A(16x4 f32) x B(4x16 f32) + C(16x16 f32)
    return __builtin_amdgcn_wmma_f32_16x16x4_f32(
        /*neg_a=*/false, a, /*neg_b=*/false, b,
        /*c_mod=*/(short)0, c, /*reuse_a=*/false, /*reuse_b=*/false);
}

// ---------------------------------------------------------------------------
// Kernel 1: fused Q/K/V projections. One wave computes a 16x64 tile of
// X(2048x128) @ W^T (+bias): one A fragment reused across 4 B tiles.
// grid.x: 0..63 -> Q cols, 64..79 -> K cols, 80..95 -> V cols.
// Outputs stored [head][s][d].
// ---------------------------------------------------------------------------
__global__ __launch_bounds__(32) void gqa_proj(
    const float* __restrict__ query, const float* __restrict__ key,
    const float* __restrict__ values,
    const float* __restrict__ Wq, const float* __restrict__ bq,
    const float* __restrict__ Wk, const float* __restrict__ bk,
    const float* __restrict__ Wv, const float* __restrict__ bv,
    float* __restrict__ qb, float* __restrict__ kraw, float* __restrict__ vraw)
{
    const int lane = threadIdx.x;
    const int m    = lane & 15;          // row (A) / col (B) within tile
    const int kb   = (lane >> 4) << 1;   // K sub-offset per ISA A/B layout
    const int half = lane >> 4;
    const int s0   = blockIdx.y << 4;
    const int bx   = blockIdx.x;

    const float* X; const float* W; const float* bias; float* dst; int col;
    if (bx < 64)      { X = query;  W = Wq; bias = bq; col = bx << 6;        dst = qb;   }
    else if (bx < 80) { X = key;    W = Wk; bias = bk; col = (bx - 64) << 6; dst = kraw; }
    else              { X = values; W = Wv; bias = bv; col = (bx - 80) << 6; dst = vraw; }

    const float* xr = X + (s0 + m) * DHEAD;      // A: row s0+m
    const float* wr[4];
#pragma unroll
    for (int t = 0; t < 4; ++t)
        wr[t] = W + (col + (t << 4) + m) * DHEAD;

    v8f c[4];
    const v8f vzero = {};
#pragma unroll
    for (int t = 0; t < 4; ++t) c[t] = vzero;

#pragma unroll
    for (int k = 0; k < DHEAD; k += 4) {
        v2f a;
        a[0] = xr[k + kb]; a[1] = xr[k + kb + 1];
#pragma unroll
        for (int t = 0; t < 4; ++t) {
            v2f b;
            b[0] = wr[t][k + kb]; b[1] = wr[t][k + kb + 1];
            c[t] = wmma_f32(a, b, c[t]);
        }
    }
#pragma unroll
    for (int t = 0; t < 4; ++t) {
        const int colt = col + (t << 4);
        const float bb = bias[colt + m];
        const int h = colt / DHEAD;
        const int d = (colt & (DHEAD - 1)) + m;
#pragma unroll
        for (int r = 0; r < 8; ++r) {
            const int srow = s0 + r + (half << 3);   // C/D: VGPR r -> M = r + 8*half
            dst[((h * S_LEN) + srow) * DHEAD + d] = c[t][r] + bb;
        }
    }
}

// ---------------------------------------------------------------------------
// Kernel 2: RoPE on K and V. K written transposed kT[h][d][s] so the
// attention B-fragments are lane-contiguous over s. V written [h][s][d].
// pairs (d, d+64) within each 128-wide head.
// ---------------------------------------------------------------------------
__global__ void gqa_rope(const float* __restrict__ kraw, const float* __restrict__ vraw,
                         float* __restrict__ kT, float* __restrict__ rvb)
{
    const int tid = blockIdx.x * blockDim.x + threadIdx.x;   // 8*2048*64 threads
    const int h   = tid >> 17;
    const int rem = tid & ((1 << 17) - 1);
    const int s   = rem >> 6;
    const int d   = rem & 63;

    // inv_freq = 10000^(-d/64) = exp(-d * ln(10000)/64)
    const float inv_freq = __expf(-(float)d * (9.210340371976184f / 64.0f));
    const float ang = (float)s * inv_freq;
    float sn, cs;
    sincosf(ang, &sn, &cs);

    const int base = ((h * S_LEN) + s) * DHEAD;
    const float k1 = kraw[base + d], k2 = kraw[base + d + 64];
    const float v1 = vraw[base + d], v2 = vraw[base + d + 64];

    kT[(h * DHEAD + d)      * S_LEN + s] = k1 * cs - k2 * sn;
    kT[(h * DHEAD + d + 64) * S_LEN + s] = k1 * sn + k2 * cs;
    rvb[base + d]      = v1 * cs - v2 * sn;
    rvb[base + d + 64] = v1 * sn + v2 * cs;
}

// ---------------------------------------------------------------------------
// Kernel 3: causal flash attention, 2-way q-row blocked. One wave handles
// 32 q rows (two 16-row sub-tiles u=0,1) per (head): every kT and rv
// B-fragment feeds two WMMAs, halving the O(S^2) K/V streaming traffic.
// Online softmax; P goes through LDS to convert C/D-layout -> A-layout.
// ---------------------------------------------------------------------------
__global__ __launch_bounds__(32) void gqa_attn(
    const float* __restrict__ qb, const float* __restrict__ kT,
    const float* __restrict__ rvb, float* __restrict__ ao)
{
    const int lane = threadIdx.x;
    const int m    = lane & 15;
    const int kb   = (lane >> 4) << 1;
    const int half = lane >> 4;
    const int h    = blockIdx.y;
    const int kv   = h & (NKV - 1);          // jnp.tile: kv head = h % 8
    const int qs0  = blockIdx.x << 5;        // 32 q rows per wave

    const float* qrow[2];
#pragma unroll
    for (int u = 0; u < 2; ++u)
        qrow[u] = qb + ((h * S_LEN) + qs0 + (u << 4) + m) * DHEAD;
    const float* kTh = kT + kv * DHEAD * S_LEN;
    const float* rvh = rvb + kv * S_LEN * DHEAD;

    __shared__ float pls[2 * 256];

    float rmax[2][8], rsum[2][8];
    v8f acc[2][8];
    const v8f vzero = {};
#pragma unroll
    for (int u = 0; u < 2; ++u)
#pragma unroll
        for (int r = 0; r < 8; ++r) {
            rmax[u][r] = -__builtin_inff(); rsum[u][r] = 0.0f; acc[u][r] = vzero;
        }

    const float scale = 0.08838834764831845f;    // 1/sqrt(128)
    const int jmax = (qs0 >> 4) + 1;             // causal tile limit for 32 rows

    for (int j = 0; j <= jmax; ++j) {
        const int ks0 = j << 4;

        // S = Q(32x128) @ K^T(128x16): 64 fp32 WMMAs, kT fragment reused 2x
        v8f sc[2];
        sc[0] = vzero; sc[1] = vzero;
#pragma unroll
        for (int k = 0; k < DHEAD; k += 4) {
            v2f b;
            b[0] = kTh[(k + kb) * S_LEN + ks0 + m];
            b[1] = kTh[(k + kb + 1) * S_LEN + ks0 + m];
#pragma unroll
            for (int u = 0; u < 2; ++u) {
                v2f a;
                a[0] = qrow[u][k + kb]; a[1] = qrow[u][k + kb + 1];
                sc[u] = wmma_f32(a, b, sc[u]);
            }
        }

        // scale + causal mask + online softmax row stats (per 16-lane half)
        float corr[2][8];
#pragma unroll
        for (int u = 0; u < 2; ++u)
#pragma unroll
            for (int r = 0; r < 8; ++r) {
                const int qi = qs0 + (u << 4) + r + (half << 3);  // global q row
                const int ki = ks0 + m;                           // global k col
                float s = sc[u][r] * scale;
                s = (ki > qi) ? -__builtin_inff() : s;

                float mx = s;
                mx = fmaxf(mx, __shfl_xor(mx, 1, 32));
                mx = fmaxf(mx, __shfl_xor(mx, 2, 32));
                mx = fmaxf(mx, __shfl_xor(mx, 4, 32));
                mx = fmaxf(mx, __shfl_xor(mx, 8, 32));
                const float nm = fmaxf(rmax[u][r], mx);  // finite after 1st tile

                const float p = __expf(s - nm);
                float sum = p;
                sum += __shfl_xor(sum, 1, 32);
                sum += __shfl_xor(sum, 2, 32);
                sum += __shfl_xor(sum, 4, 32);
                sum += __shfl_xor(sum, 8, 32);

                corr[u][r] = __expf(rmax[u][r] - nm);
                rsum[u][r] = rsum[u][r] * corr[u][r] + sum;
                rmax[u][r] = nm;
                sc[u][r]   = p;
            }
#pragma unroll
        for (int u = 0; u < 2; ++u)
#pragma unroll
            for (int t = 0; t < 8; ++t)
#pragma unroll
                for (int r = 0; r < 8; ++r)
                    acc[u][t][r] *= corr[u][r];

        // C/D layout -> LDS -> A layout for P (both sub-tiles)
#pragma unroll
        for (int u = 0; u < 2; ++u)
#pragma unroll
            for (int r = 0; r < 8; ++r)
                pls[(u << 8) + (r + (half << 3)) * 16 + m] = sc[u][r];
        __syncthreads();
        v2f pa[2][4];
#pragma unroll
        for (int u = 0; u < 2; ++u)
#pragma unroll
            for (int kk = 0; kk < 4; ++kk) {
                pa[u][kk][0] = pls[(u << 8) + m * 16 + kk * 4 + kb];
                pa[u][kk][1] = pls[(u << 8) + m * 16 + kk * 4 + kb + 1];
            }
        __syncthreads();

        // O += P(32x16) @ V(16x128): 64 fp32 WMMAs, rv fragment reused 2x
#pragma unroll
        for (int t = 0; t < 8; ++t) {
            const int n0 = t << 4;
#pragma unroll
            for (int kk = 0; kk < 4; ++kk) {
                v2f b;
                b[0] = rvh[(ks0 + kk * 4 + kb)     * DHEAD + n0 + m];
                b[1] = rvh[(ks0 + kk * 4 + kb + 1) * DHEAD + n0 + m];
#pragma unroll
                for (int u = 0; u < 2; ++u)
                    acc[u][t] = wmma_f32(pa[u][kk], b, acc[u][t]);
            }
        }
    }

    // normalize and write pre-Wo activation [s][h*128+d]
#pragma unroll
    for (int u = 0; u < 2; ++u)
#pragma unroll
        for (int r = 0; r < 8; ++r) {
            const float inv = 1.0f / rsum[u][r];
            const int srow = qs0 + (u << 4) + r + (half << 3);
#pragma unroll
            for (int t = 0; t < 8; ++t)
                ao[srow * (NQH * DHEAD) + h * DHEAD + (t << 4) + m] = acc[u][t][r] * inv;
        }
}

// ---------------------------------------------------------------------------
// Kernel 4: output projection out(2048x128) = ao(2048x4096) @ Wo^T + bo.
// One wave computes a full 16x128 output tile: one A fragment reused across
// all 8 Wo B tiles per K-step.
// ---------------------------------------------------------------------------
__global__ __launch_bounds__(32) void gqa_oproj(
    const float* __restrict__ ao, const float* __restrict__ Wo,
    const float* __restrict__ bo, float* __restrict__ out)
{
    const int lane = threadIdx.x;
    const int m    = lane & 15;
    const int kb   = (lane >> 4) << 1;
    const int half = lane >> 4;
    const int s0   = blockIdx.x << 4;

    const float* xr = ao + (s0 + m) * (NQH * DHEAD);
    const float* wr[8];
#pragma unroll
    for (int t = 0; t < 8; ++t)
        wr[t] = Wo + ((t << 4) + m) * (NQH * DHEAD);

    v8f c[8];
    const v8f vzero = {};
#pragma unroll
    for (int t = 0; t < 8; ++t) c[t] = vzero;

    for (int k = 0; k < NQH * DHEAD; k += 4) {
        v2f a;
        a[0] = xr[k + kb]; a[1] = xr[k + kb + 1];
#pragma unroll
        for (int t = 0; t < 8; ++t) {
            v2f b;
            b[0] = wr[t][k + kb]; b[1] = wr[t][k + kb + 1];
            c[t] = wmma_f32(a, b, c[t]);
        }
    }
#pragma unroll
    for (int t = 0; t < 8; ++t) {
        const float bb = bo[(t << 4) + m];
#pragma unroll
        for (int r = 0; r < 8; ++r) {
            const int srow = s0 + r + (half << 3);
            out[srow * DHEAD + (t << 4) + m] = c[t][r] + bb;
        }
    }
}

// ---------------------------------------------------------------------------
extern "C" void kernel_launch(void* const* d_in, const int* in_sizes, int n_in,
                              void* d_out, int out_size, void* d_ws, size_t ws_size,
                              hipStream_t stream) {
    (void)in_sizes; (void)n_in; (void)out_size; (void)ws_size;
    const float* query  = (const float*)d_in[0];
    const float* key    = (const float*)d_in[1];
    const float* values = (const float*)d_in[2];
    // d_in[3] = causal tril mask (int32); handled analytically in-kernel
    const float* Wq = (const float*)d_in[4];
    const float* bq = (const float*)d_in[5];
    const float* Wk = (const float*)d_in[6];
    const float* bk = (const float*)d_in[7];
    const float* Wv = (const float*)d_in[8];
    const float* bv = (const float*)d_in[9];
    const float* Wo = (const float*)d_in[10];
    const float* bo = (const float*)d_in[11];
    float* out = (float*)d_out;

    // Workspace layout (floats), lifetimes aliased: kraw/vraw die before ao is
    // written. Total 20,971,520 floats = 80 MB.
    float* ws   = (float*)d_ws;
    float* qb   = ws;              // 32*2048*128 = 8388608
    float* kT   = ws + 8388608;    //  8*128*2048 = 2097152 (transposed roped K)
    float* rvb  = ws + 10485760;   //  8*2048*128 = 2097152 (roped V)
    float* kraw = ws + 12582912;   //  2097152 (temp)
    float* vraw = ws + 14680064;   //  2097152 (temp)
    float* ao   = ws + 12582912;   //  2048*4096 = 8388608 (aliases kraw/vraw)

    gqa_proj <<<dim3(96, 128), 32, 0, stream>>>(query, key, values,
                                                Wq, bq, Wk, bk, Wv, bv,
                                                qb, kraw, vraw);
    gqa_rope <<<4096, 256, 0, stream>>>(kraw, vraw, kT, rvb);
    gqa_attn <<<dim3(64, 32), 32, 0, stream>>>(qb, kT, rvb, ao);
    gqa_oproj<<<128, 32, 0, stream>>>(ao, Wo, bo, out);
}